// PatchAttacker_58634893525576
// MI455X (gfx1250) — compile-verified
//
#include <hip/hip_runtime.h>
#include <hip/hip_bf16.h>
#include <math.h>

#define PB   128
#define PHW  256
#define BB   8
#define HH   1024
#define WW   1024
#define NB   16
#define TVM  0.5f
#define EPSF 1e-12f

typedef __attribute__((ext_vector_type(2))) float v2f;
typedef __attribute__((ext_vector_type(8))) float v8f;

// Per-lane bilinear tap setup for upsample row o (jax half-pixel, edge-renormalized):
// U[o][k] = (k==i0)*wlo + (k==i0+1)*whi ; zero for all other k.
__device__ __forceinline__ void utaps(int o, int& i0, float& wlo, float& whi) {
  float src = 0.5f * (float)o - 0.25f;
  float fi0 = floorf(src);
  i0 = (int)fi0;
  float f = src - fi0;                            // 0.25 or 0.75
  bool edge = (i0 < 0) || (i0 + 1 > PB - 1);      // only rows 0 and 255
  float inv = edge ? (1.0f / 0.75f) : 1.0f;       // 1/ws, no divide emitted
  wlo = (1.0f - f) * inv;                         // tap at k == i0
  whi = f * inv;                                  // tap at k == i0+1
}

// Aligned start of the (<=16 wide) nonzero U-column band for rows [m0, m0+15].
__device__ __forceinline__ int uband_start(int m0) {
  int klo = m0 / 2 - 1; if (klo < 0) klo = 0;
  return klo & ~3;
}

// K1: antialiased 2x downsample of patch (256x256x3 -> 128x128x3, interleaved)
__global__ void k_downsample(const float* __restrict__ patch, float* __restrict__ small) {
  int idx = blockIdx.x * blockDim.x + threadIdx.x;
  if (idx >= PB * PB * 3) return;
  int c = idx % 3, x = (idx / 3) % PB, y = idx / (3 * PB);
  const float wt[4] = {0.25f, 0.75f, 0.75f, 0.25f};
  float wx_v[4]; int xs[4];
  float wx_s = 0.f, wy_s = 0.f, sum = 0.f;
  for (int t = 0; t < 4; t++) {
    int sx = 2 * x - 1 + t;
    bool v = (sx >= 0) && (sx < PHW);
    xs[t] = v ? sx : 0;
    wx_v[t] = v ? wt[t] : 0.f;
    wx_s += wx_v[t];
  }
  for (int t = 0; t < 4; t++) {
    int sy = 2 * y - 1 + t;
    bool v = (sy >= 0) && (sy < PHW);
    float wy = v ? wt[t] : 0.f;
    wy_s += wy;
    if (wy != 0.f) {
      float row = 0.f;
      for (int u = 0; u < 4; u++) row += wx_v[u] * patch[(sy * PHW + xs[u]) * 3 + c];
      sum += wy * row;
    }
  }
  small[idx] = sum / (wy_s * wx_s);
}

// K2a: bulk float4 copy images -> out
__global__ void k_copy(const float4* __restrict__ src, float4* __restrict__ dst, int n4) {
  int i = blockIdx.x * blockDim.x + threadIdx.x;
  if (i < n4) dst[i] = src[i];
}

// K2b: paste transformed small into out; skip pixels covered by a later box (scan order)
__global__ void k_paste(const float* __restrict__ small, const int* __restrict__ box_yx,
                        const int* __restrict__ dec, float* __restrict__ out) {
  int b = blockIdx.z, n = blockIdx.y;
  int t = blockIdx.x * blockDim.x + threadIdx.x;   // 0..16383
  int i = t >> 7, j = t & (PB - 1);
  int y0 = box_yx[(b * NB + n) * 2 + 0];
  int x0 = box_yx[(b * NB + n) * 2 + 1];
  int y = y0 + i, x = x0 + j;
  for (int m = n + 1; m < NB; m++) {
    int ym = box_yx[(b * NB + m) * 2 + 0];
    int xm = box_yx[(b * NB + m) * 2 + 1];
    if (y >= ym && y < ym + PB && x >= xm && x < xm + PB) return; // later box wins
  }
  int d0 = dec[(b * NB + n) * 3 + 0];
  int d1 = dec[(b * NB + n) * 3 + 1];
  int d2 = dec[(b * NB + n) * 3 + 2];
  int a = i, bb = j;
  if (d2 > 0) a = PB - 1 - a;                 // flip_up_down (applied last in fwd)
  if (d1 > 0) bb = PB - 1 - bb;               // flip_left_right
  if (d0 > 0) { int ta = a; a = bb; bb = PB - 1 - ta; }  // rot90 k=1
  const float* s = &small[(a * PB + bb) * 3];
  float* o = &out[(((size_t)b * HH + y) * WW + x) * 3];
  o[0] = s[0]; o[1] = s[1]; o[2] = s[2];
}

// K3: sum inverse-transformed gradient windows over all 128 boxes -> planar acc[c][128][128]
__global__ void k_gather(const float* __restrict__ grads, const int* __restrict__ box_yx,
                         const int* __restrict__ dec, float* __restrict__ accp) {
  int idx = blockIdx.x * blockDim.x + threadIdx.x;
  if (idx >= 3 * PB * PB) return;
  int c = idx / (PB * PB);
  int rem = idx % (PB * PB);
  int p = rem >> 7, q = rem & (PB - 1);
  float s = 0.f;
  for (int bn = 0; bn < BB * NB; bn++) {
    int y0 = box_yx[bn * 2 + 0], x0 = box_yx[bn * 2 + 1];
    int d0 = dec[bn * 3 + 0], d1 = dec[bn * 3 + 1], d2 = dec[bn * 3 + 2];
    int a = p, bb = q;
    if (d0 > 0) { a = PB - 1 - q; bb = p; }   // rot90 k=3 (applied last in inverse chain)
    if (d1 > 0) bb = PB - 1 - bb;
    if (d2 > 0) a = PB - 1 - a;
    int b = bn >> 4;
    s += grads[(((size_t)b * HH + (y0 + a)) * WW + (x0 + bb)) * 3 + c];
  }
  accp[c * PB * PB + p * PB + q] = s;
}

// K4: stage A GEMM: tmp[c](256x128) = U(256x128) x acc[c](128x128)
// Scalar tile indices (readfirstlane), fixed 4 unrolled k-steps over the nonzero band.
__global__ void __launch_bounds__(256) k_upA(const float* __restrict__ accp, float* __restrict__ tmp) {
  int wid = __builtin_amdgcn_readfirstlane((blockIdx.x * blockDim.x + threadIdx.x) >> 5); // 0..383
  int c = wid >> 7, t = wid & 127;
  int m0 = (t >> 3) * 16;   // 0..240
  int n0 = (t & 7) * 16;    // 0..112
  int lane = threadIdx.x & 31;
  int half = lane >> 4, l16 = lane & 15;

  int i0; float wlo, whi;
  utaps(m0 + l16, i0, wlo, whi);       // loop-invariant per lane
  int ks = uband_start(m0);            // SGPR

  v8f d = {};
#pragma unroll
  for (int it = 0; it < 4; it++) {
    int ka = ks + it * 4 + half * 2;   // even; up to 130 when m0=240; row 255 keeps A weight whi at ka=128
    int kb = (ka > PB - 2) ? (PB - 2) : ka;  // clamp load rows; row 255's ka=128 tap then reads row 126
    v2f a, b;
    a.x = ((ka     == i0) ? wlo : 0.f) + ((ka     == i0 + 1) ? whi : 0.f);
    a.y = ((ka + 1 == i0) ? wlo : 0.f) + ((ka + 1 == i0 + 1) ? whi : 0.f);
    b.x = accp[c * PB * PB + kb * PB + n0 + l16];
    b.y = accp[c * PB * PB + (kb + 1) * PB + n0 + l16];
    d = __builtin_amdgcn_wmma_f32_16x16x4_f32(false, a, false, b, (short)0, d, false, false);
  }
  for (int v = 0; v < 8; v++)
    tmp[c * PHW * PB + (m0 + v + half * 8) * PB + n0 + l16] = d[v];
}

// K5: stage B GEMM: agg[.,.,c](256x256) = tmp[c](256x128) x U^T(128x256)
__global__ void __launch_bounds__(256) k_upB(const float* __restrict__ tmp, float* __restrict__ agg) {
  int wid = __builtin_amdgcn_readfirstlane((blockIdx.x * blockDim.x + threadIdx.x) >> 5); // 0..767
  int c = wid >> 8, t = wid & 255;
  int m0 = (t >> 4) * 16;
  int n0 = (t & 15) * 16;
  int lane = threadIdx.x & 31;
  int half = lane >> 4, l16 = lane & 15;

  int i0; float wlo, whi;
  utaps(n0 + l16, i0, wlo, whi);       // B rows come from U columns of output col n
  int ks = uband_start(n0);            // SGPR

  v8f d = {};
#pragma unroll
  for (int it = 0; it < 4; it++) {
    int ka = ks + it * 4 + half * 2;
    int kb = (ka > PB - 2) ? (PB - 2) : ka;  // clamp A-operand loads; col 255's ka=128 tap then reads tmp col 126
    v2f a, b;
    a.x = tmp[c * PHW * PB + (m0 + l16) * PB + kb];
    a.y = tmp[c * PHW * PB + (m0 + l16) * PB + kb + 1];
    b.x = ((ka     == i0) ? wlo : 0.f) + ((ka     == i0 + 1) ? whi : 0.f);
    b.y = ((ka + 1 == i0) ? wlo : 0.f) + ((ka + 1 == i0 + 1) ? whi : 0.f);
    d = __builtin_amdgcn_wmma_f32_16x16x4_f32(false, a, false, b, (short)0, d, false, false);
  }
  for (int v = 0; v < 8; v++)
    agg[((m0 + v + half * 8) * PHW + (n0 + l16)) * 3 + c] = d[v];
}

// K6: add TV-loss gradient of patch into agg
__global__ void k_tv(const float* __restrict__ patch, float* __restrict__ agg) {
  int idx = blockIdx.x * blockDim.x + threadIdx.x;
  if (idx >= PHW * PHW * 3) return;
  int c = idx % 3, j = (idx / 3) % PHW, i = idx / (3 * PHW);
  #define PAT(y, x) patch[((y) * PHW + (x)) * 3 + c]
  float pc = PAT(i, j);
  float dx = (j < PHW - 1) ? (pc - PAT(i, j + 1)) : 0.f;
  float dy = (i < PHW - 1) ? (pc - PAT(i + 1, j)) : 0.f;
  float g = (dx + dy) / sqrtf(dx * dx + dy * dy + EPSF);
  if (j > 0) {
    float pl = PAT(i, j - 1);
    float dxl = pl - pc;
    float dyl = (i < PHW - 1) ? (pl - PAT(i + 1, j - 1)) : 0.f;
    g -= dxl / sqrtf(dxl * dxl + dyl * dyl + EPSF);
  }
  if (i > 0) {
    float pu = PAT(i - 1, j);
    float dyu = pu - pc;
    float dxu = (j < PHW - 1) ? (pu - PAT(i - 1, j + 1)) : 0.f;
    g -= dyu / sqrtf(dxu * dxu + dyu * dyu + EPSF);
  }
  #undef PAT
  agg[idx] += TVM * g;
}

extern "C" void kernel_launch(void* const* d_in, const int* in_sizes, int n_in,
                              void* d_out, int out_size, void* d_ws, size_t ws_size,
                              hipStream_t stream) {
  const float* images = (const float*)d_in[0];
  const float* grads  = (const float*)d_in[1];
  const float* patch  = (const float*)d_in[2];
  const int*   box_yx = (const int*)d_in[3];
  const int*   dec    = (const int*)d_in[4];
  float* out = (float*)d_out;
  const size_t IMG = (size_t)BB * HH * WW * 3;   // 25,165,824

  float* ws    = (float*)d_ws;
  float* accp  = ws;                 // 3*128*128 = 49152 floats
  float* small = ws + 49152;         // 49152 floats
  float* tmp   = ws + 98304;         // 3*256*128 = 98304 floats

  // forward
  k_downsample<<<(PB * PB * 3 + 255) / 256, 256, 0, stream>>>(patch, small);
  int n4 = (int)(IMG / 4);
  k_copy<<<(n4 + 255) / 256, 256, 0, stream>>>((const float4*)images, (float4*)out, n4);
  dim3 gp(64, NB, BB);
  k_paste<<<gp, 256, 0, stream>>>(small, box_yx, dec, out);

  // backward
  k_gather<<<(3 * PB * PB + 255) / 256, 256, 0, stream>>>(grads, box_yx, dec, accp);
  k_upA<<<48, 256, 0, stream>>>(accp, tmp);          // 384 waves
  k_upB<<<96, 256, 0, stream>>>(tmp, out + IMG);     // 768 waves
  k_tv<<<(PHW * PHW * 3 + 255) / 256, 256, 0, stream>>>(patch, out + IMG);
}